// GNNLayer_62053687493141
// MI455X (gfx1250) — compile-verified
//
#include <hip/hip_runtime.h>
#include <hip/hip_bf16.h>

typedef __attribute__((ext_vector_type(2))) float v2f;
typedef __attribute__((ext_vector_type(8))) float v8f;

#define D 64
#define WAVES_PER_BLOCK 8

// ---------------------------------------------------------------------------
// Kernel 1: edge gather + scatter-add (mean numerator) + degree count.
// 16 threads per edge, float4 per thread, hardware f32 atomics.
// ---------------------------------------------------------------------------
__global__ void gnn_edge_scatter(const float* __restrict__ feat,
                                 const int*   __restrict__ src,
                                 const int*   __restrict__ dst,
                                 float*       __restrict__ msum,
                                 float*       __restrict__ deg,
                                 int E) {
    int idx = blockIdx.x * blockDim.x + threadIdx.x;
    int e = idx >> 4;          // edge id
    int t = idx & 15;          // 16 threads per edge, 4 floats each
    if (e >= E) return;

    int s = src[e];
    int d = dst[e];

    const float4 v = *(const float4*)(feat + (size_t)s * D + t * 4);
    float* p = msum + (size_t)d * D + t * 4;
    unsafeAtomicAdd(p + 0, v.x);
    unsafeAtomicAdd(p + 1, v.y);
    unsafeAtomicAdd(p + 2, v.z);
    unsafeAtomicAdd(p + 3, v.w);
    if (t == 0) unsafeAtomicAdd(deg + d, 1.0f);
}

// ---------------------------------------------------------------------------
// Kernel 2: h = msum / max(deg,1); out = h @ W^T + b  via V_WMMA_F32_16X16X4_F32
// One wave (32 lanes) per 16-node tile; 8 waves per block.
// W staged in LDS with pitch 65 to avoid bank conflicts on B-fragment reads.
// ---------------------------------------------------------------------------
__global__ void gnn_linear_wmma(const float* __restrict__ msum,
                                const float* __restrict__ deg,
                                const float* __restrict__ Wmat,  // [D_OUT, D_IN] row-major
                                const float* __restrict__ bias,
                                float*       __restrict__ out,
                                int ntiles) {
    __shared__ float ldsW[D * 65];

    // cooperative stage of W (64x64 floats) into LDS, padded pitch 65
    for (int i = threadIdx.x; i < D * D; i += blockDim.x) {
        int n = i >> 6;          // output-feature row
        int k = i & 63;          // input-feature col
        ldsW[n * 65 + k] = Wmat[i];
    }
    __syncthreads();

    const int wave = threadIdx.x >> 5;
    const int lane = threadIdx.x & 31;
    const int tile = blockIdx.x * WAVES_PER_BLOCK + wave;
    if (tile >= ntiles) return;

    const int base   = tile * 16;            // first node of this tile
    const int mrow   = lane & 15;            // M index this lane owns for A
    const int node   = base + mrow;
    const int kofs   = (lane < 16) ? 0 : 2;  // K sub-offset per lane half
    const int hi8    = (lane >> 4) * 8;      // C/D: lanes 16-31 hold M=r+8

    // mean normalization factor for this lane's row
    const float dg  = deg[node];
    const float inv = 1.0f / fmaxf(dg, 1.0f);

    // Load + normalize the full A tile (16x64) fragments: 16 K-steps of float2
    float ax[16], ay[16];
    const float* hptr = msum + (size_t)node * D + kofs;
#pragma unroll
    for (int s = 0; s < 16; ++s) {
        float2 t = *(const float2*)(hptr + s * 4);
        ax[s] = t.x * inv;
        ay[s] = t.y * inv;
    }

    // 4 N-tiles of 16 output columns each
#pragma unroll
    for (int nt = 0; nt < 4; ++nt) {
        const int col = nt * 16 + mrow;      // this lane's N index (B and C/D)
        const float bv = bias[col];
        v8f c = {bv, bv, bv, bv, bv, bv, bv, bv};

#pragma unroll
        for (int s = 0; s < 16; ++s) {
            const int k = s * 4 + kofs;
            v2f a = {ax[s], ay[s]};
            // B[k][n] = W[n][k]  (B fragment mirrors A layout with N across lanes)
            v2f b = {ldsW[col * 65 + k], ldsW[col * 65 + k + 1]};
            c = __builtin_amdgcn_wmma_f32_16x16x4_f32(
                    false, a, false, b, (short)0, c, false, false);
        }

        // scatter D: VGPR r -> out[node_base + r + hi8][col]
        float* obase = out + (size_t)(base + hi8) * D + col;
#pragma unroll
        for (int r = 0; r < 8; ++r) {
            obase[(size_t)r * D] = c[r];
        }
    }
}

// ---------------------------------------------------------------------------
extern "C" void kernel_launch(void* const* d_in, const int* in_sizes, int n_in,
                              void* d_out, int out_size, void* d_ws, size_t ws_size,
                              hipStream_t stream) {
    const float* feature = (const float*)d_in[0];
    const int*   src     = (const int*)d_in[1];
    const int*   dst     = (const int*)d_in[2];
    const float* Wmat    = (const float*)d_in[3];
    const float* bias    = (const float*)d_in[4];

    const int N = in_sizes[0] / D;   // 100000
    const int E = in_sizes[1];       // 1600000

    float* msum = (float*)d_ws;                 // [N, 64] accumulator
    float* deg  = msum + (size_t)N * D;         // [N] degree
    float* out  = (float*)d_out;

    // zero the accumulators every launch (deterministic, graph-capturable)
    hipMemsetAsync(d_ws, 0, ((size_t)N * D + N) * sizeof(float), stream);

    // edge scatter: 16 threads per edge
    {
        long long total = (long long)E * 16;
        int grid = (int)((total + 255) / 256);
        gnn_edge_scatter<<<grid, 256, 0, stream>>>(feature, src, dst, msum, deg, E);
    }

    // mean + linear via WMMA: 1 wave per 16-node tile, 8 waves per block
    {
        int ntiles = (N + 15) / 16;
        int grid = (ntiles + WAVES_PER_BLOCK - 1) / WAVES_PER_BLOCK;
        gnn_linear_wmma<<<grid, 256, 0, stream>>>(msum, deg, Wmat, bias, out, ntiles);
    }
}